// CenterLoss_69681549410333
// MI455X (gfx1250) — compile-verified
//
#include <hip/hip_runtime.h>

typedef __attribute__((ext_vector_type(2))) float v2f;
typedef __attribute__((ext_vector_type(8))) float v8f;

#define BATCH        32768
#define NUM_CLASSES  1024
#define FEATURE_DIM  256
#define NUM_TILES    (BATCH / 16)      // 2048 tiles of 16 samples
#define BLOCKS       256
#define THREADS      256               // 8 wave32 per block
#define WAVES_PER_BLOCK (THREADS / 32)
#define WAVES_TOTAL  (BLOCKS * WAVES_PER_BLOCK)   // 2048 == NUM_TILES

// Each wave handles one 16-sample tile:
//   A = x_tile - centers[labels]   (16 x 256 f32, formed on the fly)
//   Gram = A * A^T  accumulated with v_wmma_f32_16x16x4_f32 over K
//   diag(Gram)[i] = ||x_i - c_{l_i}||^2  -> clamp -> per-wave partial sum
__global__ __launch_bounds__(THREADS) void center_loss_partial_kernel(
    const float* __restrict__ x,
    const float* __restrict__ centers,
    const int*   __restrict__ labels,
    float*       __restrict__ partials) {
  const int lane      = threadIdx.x & 31;
  const int wave      = blockIdx.x * WAVES_PER_BLOCK + (threadIdx.x >> 5);
  const int rowInTile = lane & 15;   // lanes L and L+16 share row L
  const int half      = lane >> 4;   // 0 -> K=0,1 ; 1 -> K=2,3 (A layout)

  float laneSum = 0.0f;

  for (int tile = wave; tile < NUM_TILES; tile += WAVES_TOTAL) {
    const int row = tile * 16 + rowInTile;
    const int lab = labels[row];

    const float2* __restrict__ xr =
        reinterpret_cast<const float2*>(x + (size_t)row * FEATURE_DIM);
    const float2* __restrict__ cr =
        reinterpret_cast<const float2*>(centers + (size_t)lab * FEATURE_DIM);

    v8f acc = {};
#pragma unroll 4
    for (int k0 = 0; k0 < FEATURE_DIM; k0 += 4) {
      const int idx = (k0 >> 1) + half;     // float2 index: (k0 + 2*half)/2
      const float2 xv = xr[idx];
      const float2 cv = cr[idx];
      v2f a;
      a.x = xv.x - cv.x;                    // K = k0 + 2*half
      a.y = xv.y - cv.y;                    // K = k0 + 2*half + 1
      // A (16x4) and B=A^T (4x16) are register-identical for f32 layouts:
      // D = A*A^T + C
      acc = __builtin_amdgcn_wmma_f32_16x16x4_f32(
          /*neg_a=*/false, a, /*neg_b=*/false, a,
          /*c_mod=*/(short)0, acc, /*reuse_a=*/false, /*reuse_b=*/false);
    }

    // Diagonal extraction: (i,i) -> component i @ lane i (i<8),
    //                               component i-8 @ lane i+16 (i>=8); lane 16+c reads (8+c,c).
    float d = 0.0f;
#pragma unroll
    for (int c = 0; c < 8; ++c) {
      if (lane == c)      d = acc[c];
      if (lane == 16 + c) d = acc[c];
    }
    if ((lane & 15) < 8) {                 // lanes 0..7: (c,c); lanes 16..23: (8+c,c)
      d = fminf(fmaxf(d, 1e-12f), 1e12f);  // faithful per-sample clamp
      laneSum += d;
    }
  }

  // wave32 butterfly reduction (deterministic)
#pragma unroll
  for (int off = 16; off > 0; off >>= 1)
    laneSum += __shfl_xor(laneSum, off, 32);

  if (lane == 0) partials[wave] = laneSum;
}

__global__ __launch_bounds__(256) void center_loss_finalize_kernel(
    const float* __restrict__ partials, float* __restrict__ out) {
  __shared__ float sm[256];
  float s = 0.0f;
  // fixed-order accumulation -> bitwise deterministic
  for (int i = threadIdx.x; i < WAVES_TOTAL; i += 256) s += partials[i];
  sm[threadIdx.x] = s;
  __syncthreads();
#pragma unroll
  for (int stride = 128; stride > 0; stride >>= 1) {
    if (threadIdx.x < stride) sm[threadIdx.x] += sm[threadIdx.x + stride];
    __syncthreads();
  }
  if (threadIdx.x == 0) {
    // masked zeros (C-1 per row) clamp to 1e-12 and survive the sum
    out[0] = sm[0] * (1.0f / (float)BATCH)
           + (float)(NUM_CLASSES - 1) * 1e-12f;
  }
}

extern "C" void kernel_launch(void* const* d_in, const int* in_sizes, int n_in,
                              void* d_out, int out_size, void* d_ws, size_t ws_size,
                              hipStream_t stream) {
  const float* x       = (const float*)d_in[0];   // [32768, 256] f32
  const float* centers = (const float*)d_in[1];   // [1024, 256]  f32
  const int*   labels  = (const int*)d_in[2];     // [32768] int
  float* out      = (float*)d_out;                // scalar loss
  float* partials = (float*)d_ws;                 // 2048 floats (8 KB)

  center_loss_partial_kernel<<<BLOCKS, THREADS, 0, stream>>>(
      x, centers, labels, partials);
  center_loss_finalize_kernel<<<1, 256, 0, stream>>>(partials, out);
}